// Generator_Mamba_29025388986639
// MI455X (gfx1250) — compile-verified
//
#include <hip/hip_runtime.h>
#include <hip/hip_bf16.h>
#include <math.h>

typedef __attribute__((ext_vector_type(2))) float v2f;
typedef __attribute__((ext_vector_type(8))) float v8f;

#define B_    2
#define L_    2048
#define DIN_  128
#define DM_   256
#define E_    512
#define N_    16
#define K_    4
#define NL_   4
#define R_    16
#define DOUT_ 128
#define M_    (B_ * L_)   // 4096 rows

// chunked-scan geometry: L_ = CH_ * T_
#define CH_   64          // chunks along L
#define T_    32          // timesteps per chunk

// ---------------------------------------------------------------------------
// Generic fp32 WMMA GEMM: C[M,N] = A[M,Kd] * Bw[Kd,N] (+bias) (+activation)
// One wave32 per 16x16 output tile; V_WMMA_F32_16X16X4_F32 accumulate chain.
// act: 0=none, 1=softplus.  storeT: store transposed as [batch, N, Lr].
// ---------------------------------------------------------------------------
__device__ __forceinline__ float softplus_f(float x) {
    return (x > 20.f) ? x : log1pf(expf(x));
}

__global__ __launch_bounds__(128)
void gemm_f32_wmma(const float* __restrict__ A, const float* __restrict__ Bw,
                   const float* __restrict__ bias, float* __restrict__ C,
                   int M, int N, int Kd, int lda, int ldb, int ldc,
                   int act, int storeT, int Lr)
{
    const int lane   = threadIdx.x & 31;
    const int wave   = threadIdx.x >> 5;
    const int tilesN = N >> 4;
    const int tile   = blockIdx.x * (blockDim.x >> 5) + wave;
    const int total  = (M >> 4) * tilesN;
    if (tile >= total) return;                 // wave-uniform: EXEC stays full for WMMA

    const int tm   = (tile / tilesN) << 4;
    const int tn   = (tile % tilesN) << 4;
    const int half = lane >> 4;                // ISA 32-bit A/B layout: lanes 0-15 K={0,1}, 16-31 K={2,3}
    const int lid  = lane & 15;

    const float* Arow = A  + (size_t)(tm + lid) * lda;   // lane holds row m = tm+lid of A
    const float* Bcol = Bw + (tn + lid);                 // lane holds col n = tn+lid of B

    v8f acc = {};
    for (int k0 = 0; k0 < Kd; k0 += 4) {
        const int ka = k0 + (half << 1);
        v2f a, b;
        a.x = Arow[ka];
        a.y = Arow[ka + 1];
        b.x = Bcol[(size_t)ka * ldb];
        b.y = Bcol[(size_t)(ka + 1) * ldb];
        // D = A(16x4, f32) * B(4x16, f32) + C(16x16, f32)
        acc = __builtin_amdgcn_wmma_f32_16x16x4_f32(
                  /*neg_a=*/false, a, /*neg_b=*/false, b,
                  /*c_mod=*/(short)0, acc, /*reuse_a=*/false, /*reuse_b=*/false);
    }

    const int   col = tn + lid;
    const float bv  = bias ? bias[col] : 0.f;
#pragma unroll
    for (int j = 0; j < 8; ++j) {
        const int row = tm + (half ? (j + 8) : j);   // C/D layout: VGPR j -> M=j (lo half) / j+8 (hi half)
        float v = acc[j] + bv;
        if (act == 1) v = softplus_f(v);
        if (storeT) {
            const int bb = row / Lr, ll = row % Lr;
            C[((size_t)bb * N + col) * Lr + ll] = v;
        } else {
            C[(size_t)row * ldc + col] = v;
        }
    }
}

// ---------------------------------------------------------------------------
// LayerNorm over last dim (DM_=256): one wave32 per row, shfl-tree reductions.
// ---------------------------------------------------------------------------
__global__ __launch_bounds__(256)
void layernorm_k(const float* __restrict__ h, const float* __restrict__ g,
                 const float* __restrict__ beta, float* __restrict__ out)
{
    const int lane = threadIdx.x & 31;
    const int wave = threadIdx.x >> 5;
    const int row  = blockIdx.x * (blockDim.x >> 5) + wave;
    if (row >= M_) return;

    const float* hr = h + (size_t)row * DM_;
    float vals[DM_ / 32];
    float s = 0.f;
#pragma unroll
    for (int j = 0; j < DM_ / 32; ++j) { vals[j] = hr[lane + 32 * j]; s += vals[j]; }
#pragma unroll
    for (int m = 16; m >= 1; m >>= 1) s += __shfl_xor(s, m, 32);
    const float mu = s * (1.f / DM_);

    float v = 0.f;
#pragma unroll
    for (int j = 0; j < DM_ / 32; ++j) { const float d = vals[j] - mu; v += d * d; }
#pragma unroll
    for (int m = 16; m >= 1; m >>= 1) v += __shfl_xor(v, m, 32);
    const float inv = rsqrtf(v * (1.f / DM_) + 1e-5f);

    float* orow = out + (size_t)row * DM_;
#pragma unroll
    for (int j = 0; j < DM_ / 32; ++j) {
        const int c = lane + 32 * j;
        orow[c] = (vals[j] - mu) * inv * g[c] + beta[c];
    }
}

// ---------------------------------------------------------------------------
// Depthwise causal conv (K=4) + bias + SiLU over the xc half of xz.
// ---------------------------------------------------------------------------
__global__ __launch_bounds__(256)
void conv_silu_k(const float* __restrict__ xz, const float* __restrict__ cw,
                 const float* __restrict__ cb, float* __restrict__ xc)
{
    const int idx = blockIdx.x * blockDim.x + threadIdx.x;
    if (idx >= M_ * E_) return;
    const int e  = idx % E_;
    const int bl = idx / E_;
    const int l  = bl % L_;

    float acc = cb[e];
#pragma unroll
    for (int k = 0; k < K_; ++k) {
        const int ls = l + k - (K_ - 1);
        if (ls >= 0)
            acc += cw[k * E_ + e] * xz[(size_t)(bl + ls - l) * (2 * E_) + e];
    }
    xc[idx] = acc / (1.f + expf(-acc));   // SiLU
}

// ---------------------------------------------------------------------------
// Chunked associative scan for s[t] = a[t]*s[t-1] + u[t]:
//   a[t] = exp(dt[b,t,e] * A[e,n]),  u[t] = dt*xc*Bm
// Pass 1: per (chain, chunk) compute coefficient product P_c and local sum S_c.
// Pass 2: per chain, sequentially propagate carries across the 64 chunks.
// Pass 3: re-scan each chunk seeded with its carry; fused einsum reduction
//         over n (16-lane shfl tree), D-skip, and SiLU(z) gate.
// Lane layout (passes 1/3): lane = 16*half + n; wave handles 2 (b,e) pairs.
// ---------------------------------------------------------------------------
__global__ __launch_bounds__(256)
void scan_pass1(const float* __restrict__ dt, const float* __restrict__ xc,
                const float* __restrict__ dbl, const float* __restrict__ A_log,
                float* __restrict__ P, float* __restrict__ S)
{
    const int lane = threadIdx.x & 31;
    const int gw   = (blockIdx.x * blockDim.x + threadIdx.x) >> 5;
    const int half = lane >> 4;
    const int n    = lane & 15;
    const int c    = gw & (CH_ - 1);             // chunk id
    const int p    = (gw >> 6) * 2 + half;       // (b,e) pair id
    if (p >= B_ * E_) return;
    const int b = p / E_;
    const int e = p % E_;

    const float Av = -expf(A_log[(size_t)e * N_ + n]);

    const size_t base = (size_t)b * L_ + (size_t)c * T_;
    const float* dtr = dt  + base * E_ + e;
    const float* xcr = xc  + base * E_ + e;
    const float* dr  = dbl + base * 48;

    float st = 0.f, Pp = 1.f;
#pragma unroll 4
    for (int t = 0; t < T_; ++t) {
        const float dtv = dtr[(size_t)t * E_];
        const float xcv = xcr[(size_t)t * E_];
        const float Bv  = dr[(size_t)t * 48 + R_ + n];
        const float a   = expf(dtv * Av);
        st = a * st + dtv * xcv * Bv;
        Pp *= a;
    }
    const size_t chain = (size_t)p * N_ + n;
    P[chain * CH_ + c] = Pp;
    S[chain * CH_ + c] = st;
}

__global__ __launch_bounds__(256)
void scan_pass2(const float* __restrict__ P, const float* __restrict__ S,
                float* __restrict__ carry)
{
    const int q = blockIdx.x * blockDim.x + threadIdx.x;   // chain id
    if (q >= B_ * E_ * N_) return;
    const float* Pq = P + (size_t)q * CH_;
    const float* Sq = S + (size_t)q * CH_;
    float*       Cq = carry + (size_t)q * CH_;
    float cv = 0.f;
#pragma unroll 4
    for (int c = 0; c < CH_; ++c) {
        Cq[c] = cv;                 // state entering chunk c
        cv = Pq[c] * cv + Sq[c];
    }
}

__global__ __launch_bounds__(256)
void scan_pass3(const float* __restrict__ dt, const float* __restrict__ xc,
                const float* __restrict__ dbl, const float* __restrict__ xz,
                const float* __restrict__ A_log, const float* __restrict__ D_skip,
                const float* __restrict__ carry, float* __restrict__ y)
{
    const int lane = threadIdx.x & 31;
    const int gw   = (blockIdx.x * blockDim.x + threadIdx.x) >> 5;
    const int half = lane >> 4;
    const int n    = lane & 15;
    const int c    = gw & (CH_ - 1);
    const int p    = (gw >> 6) * 2 + half;
    if (p >= B_ * E_) return;
    const int b = p / E_;
    const int e = p % E_;

    const float Av = -expf(A_log[(size_t)e * N_ + n]);
    const float Dv = D_skip[e];

    const size_t base = (size_t)b * L_ + (size_t)c * T_;
    const float* dtr = dt  + base * E_ + e;
    const float* xcr = xc  + base * E_ + e;
    const float* zr  = xz  + base * (2 * E_) + E_ + e;
    const float* dr  = dbl + base * 48;
    float*       yr  = y   + base * E_ + e;

    float st = carry[((size_t)p * N_ + n) * CH_ + c];
#pragma unroll 4
    for (int t = 0; t < T_; ++t) {
        const float dtv = dtr[(size_t)t * E_];
        const float xcv = xcr[(size_t)t * E_];
        const float Bv  = dr[(size_t)t * 48 + R_ + n];
        const float Cv  = dr[(size_t)t * 48 + R_ + N_ + n];

        const float a = expf(dtv * Av);
        st = a * st + dtv * xcv * Bv;

        float acc = st * Cv;
        acc += __shfl_xor(acc, 1, 32);
        acc += __shfl_xor(acc, 2, 32);
        acc += __shfl_xor(acc, 4, 32);
        acc += __shfl_xor(acc, 8, 32);          // 16-lane tree: stays inside each half

        if (n == 0) {
            const float zv = zr[(size_t)t * (2 * E_)];
            const float gate = zv / (1.f + expf(-zv));   // SiLU(z)
            yr[(size_t)t * E_] = (acc + Dv * xcv) * gate;
        }
    }
}

// ---------------------------------------------------------------------------
// Host-side orchestration (graph-capture safe: launches only).
// ---------------------------------------------------------------------------
extern "C" void kernel_launch(void* const* d_in, const int* in_sizes, int n_in,
                              void* d_out, int out_size, void* d_ws, size_t ws_size,
                              hipStream_t stream)
{
    const float* x      = (const float*)d_in[0];
    const float* W_ip   = (const float*)d_in[1];
    const float* b_ip   = (const float*)d_in[2];
    const float* ln_g   = (const float*)d_in[3];
    const float* ln_b   = (const float*)d_in[4];
    const float* W_in   = (const float*)d_in[5];
    const float* conv_w = (const float*)d_in[6];
    const float* conv_b = (const float*)d_in[7];
    const float* W_x    = (const float*)d_in[8];
    const float* W_dt   = (const float*)d_in[9];
    const float* b_dt   = (const float*)d_in[10];
    const float* A_log  = (const float*)d_in[11];
    const float* D_skip = (const float*)d_in[12];
    const float* W_out  = (const float*)d_in[13];
    const float* W_fc   = (const float*)d_in[14];
    const float* b_fc   = (const float*)d_in[15];
    float* out = (float*)d_out;

    // workspace carve-up (~63 MB of fp32)
    float* ws  = (float*)d_ws;
    float* h   = ws;  ws += (size_t)M_ * DM_;
    float* hn  = ws;  ws += (size_t)M_ * DM_;
    float* xz  = ws;  ws += (size_t)M_ * 2 * E_;
    float* xc  = ws;  ws += (size_t)M_ * E_;
    float* dbl = ws;  ws += (size_t)M_ * 48;
    float* dt  = ws;  ws += (size_t)M_ * E_;
    float* y   = ws;  ws += (size_t)M_ * E_;
    float* Pb  = ws;  ws += (size_t)B_ * E_ * N_ * CH_;   // chunk coefficient products
    float* Sb  = ws;  ws += (size_t)B_ * E_ * N_ * CH_;   // chunk local sums
    float* Cb  = ws;  ws += (size_t)B_ * E_ * N_ * CH_;   // chunk entry carries

    auto gemm = [&](const float* A, const float* Bw, const float* bias, float* C,
                    int M, int N, int Kd, int lda, int ldb, int ldc,
                    int act, int storeT) {
        const int tiles  = (M / 16) * (N / 16);
        const int blocks = (tiles + 3) / 4;     // 4 waves per 128-thread block
        gemm_f32_wmma<<<blocks, 128, 0, stream>>>(A, Bw, bias, C, M, N, Kd,
                                                  lda, ldb, ldc, act, storeT, L_);
    };

    const int scanWaves  = (B_ * E_ / 2) * CH_;          // 32768 waves
    const int scanBlocks = scanWaves * 32 / 256;         // 4096 blocks

    // h = x @ W_ip + b_ip
    gemm(x, W_ip, b_ip, h, M_, DM_, DIN_, DIN_, DM_, DM_, 0, 0);

    for (int i = 0; i < NL_; ++i) {
        const float* Al = A_log  + (size_t)i * E_ * N_;
        const float* Dp = D_skip + (size_t)i * E_;
        // hn = LayerNorm(h)
        layernorm_k<<<M_ / 8, 256, 0, stream>>>(h, ln_g, ln_b, hn);
        // xz = hn @ W_in[i]
        gemm(hn, W_in + (size_t)i * DM_ * 2 * E_, nullptr, xz,
             M_, 2 * E_, DM_, DM_, 2 * E_, 2 * E_, 0, 0);
        // xc = SiLU(causal_conv(xz[..., :E]) + cb)
        conv_silu_k<<<(M_ * E_ + 255) / 256, 256, 0, stream>>>(
            xz, conv_w + (size_t)i * K_ * E_, conv_b + (size_t)i * E_, xc);
        // dbl = xc @ W_x[i]   (dt_raw | B | C)
        gemm(xc, W_x + (size_t)i * E_ * (R_ + 2 * N_), nullptr, dbl,
             M_, R_ + 2 * N_, E_, E_, R_ + 2 * N_, R_ + 2 * N_, 0, 0);
        // dt = softplus(dbl[:, :R] @ W_dt[i] + b_dt[i])
        gemm(dbl, W_dt + (size_t)i * R_ * E_, b_dt + (size_t)i * E_, dt,
             M_, E_, R_, R_ + 2 * N_, E_, E_, 1, 0);
        // chunked selective scan -> y (fused D-skip + SiLU(z) gate)
        scan_pass1<<<scanBlocks, 256, 0, stream>>>(dt, xc, dbl, Al, Pb, Sb);
        scan_pass2<<<(B_ * E_ * N_) / 256, 256, 0, stream>>>(Pb, Sb, Cb);
        scan_pass3<<<scanBlocks, 256, 0, stream>>>(dt, xc, dbl, xz, Al, Dp, Cb, y);
        // h = y @ W_out[i]
        gemm(y, W_out + (size_t)i * E_ * DM_, nullptr, h,
             M_, DM_, E_, E_, DM_, DM_, 0, 0);
    }

    // out[b, d, l] = (h @ W_fc + b_fc)[b, l, d]
    gemm(h, W_fc, b_fc, out, M_, DOUT_, DM_, DM_, DOUT_, DOUT_, 0, 1);
}